// TOP_Operator_32057635897935
// MI455X (gfx1250) — compile-verified
//
#include <hip/hip_runtime.h>
#include <math.h>

// ---------------------------------------------------------------------------
// Problem constants (from reference): x [N=32, C=256, H=56, W=56] fp32
// ---------------------------------------------------------------------------
#define Ncfg 32
#define Ccfg 256
#define Hcfg 56
#define Wcfg 56
#define PLANE (Hcfg * Wcfg)       // 3136 floats, contiguous per (n,c)
#define PLANE4 (PLANE / 4)        // 784 float4
#define NTHREADS 256
#define NHW_COUNT (Ncfg * PLANE)  // 100352 elements per channel
#define BN_EPS 1e-5f

#ifndef __has_builtin
#define __has_builtin(x) 0
#endif

// Builtin signature (from hipcc diagnostic):
//   void __builtin_amdgcn_global_load_async_to_lds_b128(
//       v4i addrspace(1)* src, v4i addrspace(3)* dst, imm offset, imm cpol)
typedef int v4i __attribute__((vector_size(4 * sizeof(int))));
typedef __attribute__((address_space(1))) v4i glb_v4i;
typedef __attribute__((address_space(3))) v4i lds_v4i;
typedef __attribute__((address_space(3))) float lds_f32;

#if __has_builtin(__builtin_amdgcn_global_load_async_to_lds_b128) && \
    __has_builtin(__builtin_amdgcn_s_wait_asynccnt)
#define ASYNC_TIER 1   // clang builtin path (active for gfx1250 device pass)
#elif defined(__HIP_DEVICE_COMPILE__)
#define ASYNC_TIER 2   // device pass without builtin: inline-asm path
#else
#define ASYNC_TIER 0   // host sema pass: plain copy (never executed)
#endif

// Stage one contiguous 56x56 plane (12544 B) from global into LDS using the
// CDNA5 async global->LDS DMA path (tracked by ASYNCcnt, waited with
// s_wait_asynccnt), then rendezvous the workgroup.
__device__ __forceinline__ void load_plane_to_lds(float* smem, const float* gsrc) {
    const int tid = threadIdx.x;
#if ASYNC_TIER == 1
    for (int i = tid; i < PLANE4; i += NTHREADS) {
        __builtin_amdgcn_global_load_async_to_lds_b128(
            (glb_v4i*)(gsrc + 4 * i),
            (lds_v4i*)(smem + 4 * i),
            0, 0);
    }
    __builtin_amdgcn_s_wait_asynccnt(0);
#elif ASYNC_TIER == 2
    for (int i = tid; i < PLANE4; i += NTHREADS) {
        unsigned lds_off =
            (unsigned)(unsigned long long)(lds_f32*)(smem + 4 * i);
        unsigned long long gaddr = (unsigned long long)(gsrc + 4 * i);
        asm volatile("global_load_async_to_lds_b128 %0, %1, off"
                     :: "v"(lds_off), "v"(gaddr)
                     : "memory");
    }
    asm volatile("s_wait_asynccnt 0" ::: "memory");
#else
    const float4* s4 = (const float4*)gsrc;
    float4* d4 = (float4*)smem;
    for (int i = tid; i < PLANE4; i += NTHREADS) d4[i] = s4[i];
#endif
    __syncthreads();
}

// Compute the three depthwise conv taps for pixel p of a 56x56 LDS tile.
// (cross-correlation, zero padding — matches lax.conv_general_dilated)
__device__ __forceinline__ void conv3taps(const float* smem, int p,
                                          const float wxy[9], const float wxz[3],
                                          const float wyz[3],
                                          float& cxy, float& cxz, float& cyz) {
    const int h = p / Wcfg;
    const int w = p - h * Wcfg;
    float v[3][3];
#pragma unroll
    for (int dy = 0; dy < 3; ++dy) {
        const int hh = h + dy - 1;
        const bool hv = ((unsigned)hh < (unsigned)Hcfg);
#pragma unroll
        for (int dx = 0; dx < 3; ++dx) {
            const int ww = w + dx - 1;
            const bool wv = ((unsigned)ww < (unsigned)Wcfg);
            v[dy][dx] = (hv && wv) ? smem[hh * Wcfg + ww] : 0.0f;
        }
    }
    float a = 0.0f;
#pragma unroll
    for (int dy = 0; dy < 3; ++dy)
#pragma unroll
        for (int dx = 0; dx < 3; ++dx)
            a = fmaf(v[dy][dx], wxy[dy * 3 + dx], a);
    cxy = a;
    cxz = fmaf(v[1][0], wxz[0], fmaf(v[1][1], wxz[1], v[1][2] * wxz[2]));
    cyz = fmaf(v[0][1], wyz[0], fmaf(v[1][1], wyz[1], v[2][1] * wyz[2]));
}

// ---------------------------------------------------------------------------
// Pass 1: per-(n,c) plane -> conv taps -> per-block {sum, sumsq} x 3 branches.
// Deterministic: fixed per-thread accumulation order + fixed LDS tree.
// ---------------------------------------------------------------------------
__global__ __launch_bounds__(NTHREADS) void dw_stats_pass(
    const float* __restrict__ x,
    const float* __restrict__ w_xy,
    const float* __restrict__ w_xz,
    const float* __restrict__ w_yz,
    float* __restrict__ partials) {
    __shared__ float smem[PLANE];  // reused as reduction scratch after taps
    const int blk = blockIdx.x;    // = n*C + c  (matches NCHW plane order)
    const int c = blk % Ccfg;
    const int n = blk / Ccfg;

    load_plane_to_lds(smem, x + (size_t)blk * PLANE);

    float wxy[9], wxz[3], wyz[3];
#pragma unroll
    for (int i = 0; i < 9; ++i) wxy[i] = w_xy[c * 9 + i];
#pragma unroll
    for (int i = 0; i < 3; ++i) wxz[i] = w_xz[c * 3 + i];
#pragma unroll
    for (int i = 0; i < 3; ++i) wyz[i] = w_yz[c * 3 + i];

    float s[6] = {0.f, 0.f, 0.f, 0.f, 0.f, 0.f};
    for (int p = threadIdx.x; p < PLANE; p += NTHREADS) {
        float a, b, d;
        conv3taps(smem, p, wxy, wxz, wyz, a, b, d);
        s[0] += a; s[1] += a * a;
        s[2] += b; s[3] += b * b;
        s[4] += d; s[5] += d * d;
    }
    __syncthreads();  // everyone done reading the tile; reuse smem

#pragma unroll
    for (int k = 0; k < 6; ++k) smem[threadIdx.x + k * NTHREADS] = s[k];
    __syncthreads();
    for (int stride = NTHREADS / 2; stride > 0; stride >>= 1) {
        if ((int)threadIdx.x < stride) {
#pragma unroll
            for (int k = 0; k < 6; ++k)
                smem[threadIdx.x + k * NTHREADS] +=
                    smem[threadIdx.x + stride + k * NTHREADS];
        }
        __syncthreads();
    }
    if (threadIdx.x == 0) {
        float* dst = partials + ((size_t)c * Ncfg + n) * 6;
#pragma unroll
        for (int k = 0; k < 6; ++k) dst[k] = smem[k * NTHREADS];
    }
}

// ---------------------------------------------------------------------------
// Pass 1b: fold 32 partials per channel (fixed order) -> scale/shift per
// branch. 256 threads, thread == channel.
// ---------------------------------------------------------------------------
__global__ __launch_bounds__(Ccfg) void stats_finalize(
    const float* __restrict__ partials,
    const float* __restrict__ g_xy, const float* __restrict__ b_xy,
    const float* __restrict__ g_xz, const float* __restrict__ b_xz,
    const float* __restrict__ g_yz, const float* __restrict__ b_yz,
    float* __restrict__ stats) {
    const int c = threadIdx.x;
    float s[6] = {0.f, 0.f, 0.f, 0.f, 0.f, 0.f};
    for (int n = 0; n < Ncfg; ++n) {
        const float* p = partials + ((size_t)c * Ncfg + n) * 6;
#pragma unroll
        for (int k = 0; k < 6; ++k) s[k] += p[k];
    }
    const float inv_cnt = 1.0f / (float)NHW_COUNT;

    // branch 0: xy
    {
        float mean = s[0] * inv_cnt;
        float var = fmaxf(s[1] * inv_cnt - mean * mean, 0.0f);
        float inv = rsqrtf(var + BN_EPS);
        float sc = g_xy[c] * inv;
        stats[c * 6 + 0] = sc;
        stats[c * 6 + 1] = b_xy[c] - mean * sc;
    }
    // branch 1: xz
    {
        float mean = s[2] * inv_cnt;
        float var = fmaxf(s[3] * inv_cnt - mean * mean, 0.0f);
        float inv = rsqrtf(var + BN_EPS);
        float sc = g_xz[c] * inv;
        stats[c * 6 + 2] = sc;
        stats[c * 6 + 3] = b_xz[c] - mean * sc;
    }
    // branch 2: yz
    {
        float mean = s[4] * inv_cnt;
        float var = fmaxf(s[5] * inv_cnt - mean * mean, 0.0f);
        float inv = rsqrtf(var + BN_EPS);
        float sc = g_yz[c] * inv;
        stats[c * 6 + 4] = sc;
        stats[c * 6 + 5] = b_yz[c] - mean * sc;
    }
}

// ---------------------------------------------------------------------------
// Pass 2: recompute taps (x is L2-resident from pass 1), normalize, gate,
// ReLU, NT-store the result so output doesn't evict x from L2.
// ---------------------------------------------------------------------------
__global__ __launch_bounds__(NTHREADS) void dw_apply_pass(
    const float* __restrict__ x,
    const float* __restrict__ w_xy,
    const float* __restrict__ w_xz,
    const float* __restrict__ w_yz,
    const float* __restrict__ stats,
    float* __restrict__ out) {
    __shared__ float smem[PLANE];
    const int blk = blockIdx.x;
    const int c = blk % Ccfg;

    load_plane_to_lds(smem, x + (size_t)blk * PLANE);

    float wxy[9], wxz[3], wyz[3];
#pragma unroll
    for (int i = 0; i < 9; ++i) wxy[i] = w_xy[c * 9 + i];
#pragma unroll
    for (int i = 0; i < 3; ++i) wxz[i] = w_xz[c * 3 + i];
#pragma unroll
    for (int i = 0; i < 3; ++i) wyz[i] = w_yz[c * 3 + i];

    const float sc_xy = stats[c * 6 + 0], sh_xy = stats[c * 6 + 1];
    const float sc_xz = stats[c * 6 + 2], sh_xz = stats[c * 6 + 3];
    const float sc_yz = stats[c * 6 + 4], sh_yz = stats[c * 6 + 5];

    float* oplane = out + (size_t)blk * PLANE;
    for (int p = threadIdx.x; p < PLANE; p += NTHREADS) {
        float a, b, d;
        conv3taps(smem, p, wxy, wxz, wyz, a, b, d);
        const float fxy = fmaf(a, sc_xy, sh_xy);
        const float fxz = fmaf(b, sc_xz, sh_xz);
        const float fyz = fmaf(d, sc_yz, sh_yz);
        const float t = fxz * fyz;
        const float sig = 1.0f / (1.0f + __expf(-t));
        const float r = fmaxf(fxy * sig, 0.0f);
        __builtin_nontemporal_store(r, oplane + p);
    }
}

// ---------------------------------------------------------------------------
// Harness entry
// ---------------------------------------------------------------------------
extern "C" void kernel_launch(void* const* d_in, const int* in_sizes, int n_in,
                              void* d_out, int out_size, void* d_ws, size_t ws_size,
                              hipStream_t stream) {
    const float* x    = (const float*)d_in[0];
    const float* w_xy = (const float*)d_in[1];
    const float* w_xz = (const float*)d_in[2];
    const float* w_yz = (const float*)d_in[3];
    const float* g_xy = (const float*)d_in[4];
    const float* b_xy = (const float*)d_in[5];
    const float* g_xz = (const float*)d_in[6];
    const float* b_xz = (const float*)d_in[7];
    const float* g_yz = (const float*)d_in[8];
    const float* b_yz = (const float*)d_in[9];
    float* out = (float*)d_out;

    // Workspace layout: [partials: C*N*6 floats][stats: C*6 floats]
    float* partials = (float*)d_ws;
    float* stats = partials + (size_t)Ccfg * Ncfg * 6;

    const int nplanes = Ncfg * Ccfg;  // 8192 blocks, one 56x56 plane each
    dw_stats_pass<<<nplanes, NTHREADS, 0, stream>>>(x, w_xy, w_xz, w_yz, partials);
    stats_finalize<<<1, Ccfg, 0, stream>>>(partials, g_xy, b_xy, g_xz, b_xz,
                                           g_yz, b_yz, stats);
    dw_apply_pass<<<nplanes, NTHREADS, 0, stream>>>(x, w_xy, w_xz, w_yz, stats, out);
}